// BatchBeamSearchLayer_41987600286355
// MI455X (gfx1250) — compile-verified
//
#include <hip/hip_runtime.h>
#include <hip/hip_bf16.h>
#include <cstdint>

#define NB    128      // B
#define KB    8        // K beams
#define NV    50257    // vocab
#define LSEQ  64
#define MAXLEN 128
#define ADIM  512
#define DDIM  1024
#define EOSID 2

// ---- output layout (floats, concatenated in reference return order) ----
#define O_SCORES   0         // new_cand_scores  B*K            = 1024
#define O_NEWSEQS  1024      // new_seqs         B*K*(L+1)      = 66560
#define O_PARENTS  67584     // parents          B*K            = 1024
#define O_CSCORES  68608     // comp_scores      B*K            = 1024
#define O_CSEQS    69632     // comp_seqs        B*K*MAXL       = 131072
#define O_CLEN     200704    // comp_len         B*K            = 1024
#define O_CTX      201728    // ctx              B*K*A          = 524288
#define O_R1       726016    // r1               B*K*D          = 1048576
#define O_R2       1774592   // r2               B*K*D          = 1048576

// ---- CDNA5 async global->LDS staging (gfx1250).
// Builtin signature (from clang diagnostic): (v4i AS1*, v4i AS3*, Ii, Ii)
typedef int v4i_ __attribute__((ext_vector_type(4)));

#if __has_builtin(__builtin_amdgcn_global_load_async_to_lds_b128)
#define HAVE_ASYNC 1
#define ASYNC_B128(g, l)                                                      \
  __builtin_amdgcn_global_load_async_to_lds_b128(                             \
      (__attribute__((address_space(1))) v4i_*)(g),                           \
      (__attribute__((address_space(3))) v4i_*)(l), 0, 0)
#if __has_builtin(__builtin_amdgcn_s_wait_asynccnt)
#define WAIT_ASYNC(n) __builtin_amdgcn_s_wait_asynccnt(n)
#else
#define WAIT_ASYNC(n) asm volatile("s_wait_asynccnt %0" ::"i"(n) : "memory")
#endif
#else
#define HAVE_ASYNC 0
#endif

__device__ __forceinline__ void insert8(float x, int idx, float tv[8], int ti[8]) {
  if (x <= tv[7]) return;              // common fast reject: 1 compare/element
  float cv = x; int ci = idx;
#pragma unroll
  for (int j = 0; j < 8; ++j) {
    if (cv > tv[j]) { float tf = tv[j]; int tn = ti[j]; tv[j] = cv; ti[j] = ci; cv = tf; ci = tn; }
  }
}

// ---------------------------------------------------------------------------
// Kernel 1: fused per-(b,k)-row pass over V logits.
//   - online softmax stats (m, sum exp)
//   - per-row top-8 of raw logits with EOS excluded (monotone shift later)
//   - EOS logprob score
// One workgroup (256 thr = 8 waves, wave32) per row; async double-buffered
// LDS staging of 4KB tiles (each lane consumes exactly its own 16B slice, so
// only the per-wave ASYNCcnt wait is needed — no workgroup barrier).
// ---------------------------------------------------------------------------
__global__ __launch_bounds__(256) void k_row_fused(
    const float* __restrict__ logits, const float* __restrict__ cand_scores,
    float* __restrict__ ws_eos, float* __restrict__ ws_tv, int* __restrict__ ws_ti) {
  const int row = blockIdx.x;                 // b*K + k
  const int tid = threadIdx.x;
  const float* rowp = logits + (size_t)row * NV;

  __shared__ float4 stage[2][256];            // 8KB double buffer
  __shared__ float  sm[256], ss[256];
  __shared__ float  cval[2048];
  __shared__ int    cidx[2048];
  __shared__ float  rv[256];
  __shared__ int    rp[256];

  float m = -INFINITY, s = 0.0f;
  float tv[8]; int ti[8];
#pragma unroll
  for (int j = 0; j < 8; ++j) { tv[j] = -INFINITY; ti[j] = 0x7fffffff; }

  auto proc4 = [&](int base, float4 v) {
    float xs[4] = {v.x, v.y, v.z, v.w};
#pragma unroll
    for (int j = 0; j < 4; ++j) {
      float x = xs[j];
      if (x > m) { s = s * __expf(m - x) + 1.0f; m = x; }   // online softmax
      else       { s += __expf(x - m); }
      int gi = base + j;
      if (gi != EOSID) insert8(x, gi, tv, ti);              // EOS masked for top-k only
    }
  };

  const int TILE  = 1024;                     // floats per tile (256 lanes * 4)
  const int nfull = NV / TILE;                // 49 full tiles; tail = 81

#if HAVE_ASYNC
  ASYNC_B128(rowp + tid * 4, &stage[0][tid]);
  for (int t = 0; t < nfull; ++t) {
    if (t + 1 < nfull) {
      ASYNC_B128(rowp + (t + 1) * TILE + tid * 4, &stage[(t + 1) & 1][tid]);
      WAIT_ASYNC(1);                          // in-order: tile t is complete
    } else {
      WAIT_ASYNC(0);
    }
    proc4(t * TILE + tid * 4, stage[t & 1][tid]);
  }
#else
  for (int t = 0; t < nfull; ++t) {
    if (t + 1 < nfull) __builtin_prefetch(rowp + (t + 1) * TILE + tid * 4, 0, 0);
    proc4(t * TILE + tid * 4, *(const float4*)(rowp + t * TILE + tid * 4));
  }
#endif
  {                                           // tail (81 elements)
    int gi = nfull * TILE + tid;
    if (gi < NV) {
      float x = rowp[gi];
      if (x > m) { s = s * __expf(m - x) + 1.0f; m = x; } else { s += __expf(x - m); }
      if (gi != EOSID) insert8(x, gi, tv, ti);
    }
  }

  // ---- reduce (m, s) across 256 threads ----
  sm[tid] = m; ss[tid] = s;
  __syncthreads();
  for (int off = 128; off > 0; off >>= 1) {
    if (tid < off) {
      float m1 = sm[tid], s1 = ss[tid], m2 = sm[tid + off], s2 = ss[tid + off];
      float mm = fmaxf(m1, m2);
      sm[tid] = mm;
      ss[tid] = s1 * __expf(m1 - mm) + s2 * __expf(m2 - mm);
    }
    __syncthreads();
  }
  const float logZ = sm[0] + __logf(ss[0]);
  const float shift = cand_scores[row] - logZ;   // logprob -> beam score

  // ---- merge 256 x top8 -> row top8 (8 argmax rounds over 2048) ----
#pragma unroll
  for (int j = 0; j < 8; ++j) { cval[tid * 8 + j] = tv[j]; cidx[tid * 8 + j] = ti[j]; }
  __syncthreads();
  for (int r = 0; r < 8; ++r) {
    float best = -INFINITY; int bp = 0;
#pragma unroll
    for (int j = 0; j < 8; ++j) {
      int p = tid * 8 + j;
      if (cval[p] > best) { best = cval[p]; bp = p; }
    }
    rv[tid] = best; rp[tid] = bp;
    __syncthreads();
    for (int off = 128; off > 0; off >>= 1) {
      if (tid < off && rv[tid + off] > rv[tid]) { rv[tid] = rv[tid + off]; rp[tid] = rp[tid + off]; }
      __syncthreads();
    }
    if (tid == 0) {
      int w = rp[0];
      ws_tv[row * 8 + r] = rv[0] + shift;       // final beam-score domain
      ws_ti[row * 8 + r] = cidx[w];             // symbol (index within V)
      cval[w] = -INFINITY;
    }
    __syncthreads();
  }
  if (tid == 0) ws_eos[row] = rowp[EOSID] + shift;   // new_scores[:,:,EOS]
}

// ---------------------------------------------------------------------------
// Kernel 2: completed-hypothesis path. One block per batch.
// ---------------------------------------------------------------------------
__global__ __launch_bounds__(128) void k_completed(
    const float* __restrict__ completed_scores, const int* __restrict__ completed_seqs,
    const int* __restrict__ completed_length, const int* __restrict__ cand_seqs,
    const float* __restrict__ ws_eos, float* __restrict__ out) {
  const int b = blockIdx.x, tid = threadIdx.x;
  __shared__ float s16[16];
  __shared__ int   l16[16];
  __shared__ int   sel[8];
  if (tid < 16) {
    if (tid < 8) { s16[tid] = completed_scores[b * 8 + tid]; l16[tid] = completed_length[b * 8 + tid]; }
    else         { s16[tid] = ws_eos[b * 8 + (tid - 8)];     l16[tid] = LSEQ + 1; }
  }
  __syncthreads();
  if (tid == 0) {                              // top-8 of 16 by length-rescaled score
    unsigned used = 0;
    for (int r = 0; r < 8; ++r) {
      float best = -INFINITY; int bi = 0;
      for (int c = 0; c < 16; ++c) {
        if (used & (1u << c)) continue;
        float k = s16[c] / (float)l16[c];
        if (k > best) { best = k; bi = c; }    // strict > => first index wins ties
      }
      used |= 1u << bi; sel[r] = bi;
    }
  }
  __syncthreads();
  if (tid < 8) {
    int c = sel[tid];
    out[O_CSCORES + b * 8 + tid] = s16[c];
    out[O_CLEN    + b * 8 + tid] = (float)l16[c];
  }
  for (int r = 0; r < 8; ++r) {
    int c = sel[r];
    int dst = O_CSEQS + (b * 8 + r) * MAXLEN;
    if (c < 8) {
      const int* src = completed_seqs + ((size_t)b * 8 + c) * MAXLEN;
      for (int t = tid; t < MAXLEN; t += 128) out[dst + t] = (float)src[t];
    } else {
      const int* src = cand_seqs + ((size_t)b * 8 + (c - 8)) * LSEQ;
      for (int t = tid; t < MAXLEN; t += 128) out[dst + t] = (float)(t < LSEQ ? src[t] : EOSID);
    }
  }
}

// ---------------------------------------------------------------------------
// Kernel 3: merge 8 rows x top8 -> batch top-8 over flattened (K*V).
// ---------------------------------------------------------------------------
__global__ __launch_bounds__(128) void k_newtop(
    const float* __restrict__ ws_tv, const int* __restrict__ ws_ti,
    const int* __restrict__ cand_seqs, int* __restrict__ ws_par, float* __restrict__ out) {
  const int b = blockIdx.x, tid = threadIdx.x;
  __shared__ float cv[64];
  __shared__ int   cs[64];   // symbol (idx within V)
  __shared__ int   ck[64];   // parent row k
  __shared__ float selv[8];
  __shared__ int   selp[8], sels[8];
  if (tid < 64) {
    int k = tid >> 3, r = tid & 7;
    cv[tid] = ws_tv[(b * 8 + k) * 8 + r];
    cs[tid] = ws_ti[(b * 8 + k) * 8 + r];
    ck[tid] = k;
  }
  __syncthreads();
  if (tid == 0) {
    unsigned long long used = 0;
    for (int r = 0; r < 8; ++r) {
      float best = -INFINITY; long long bflat = 0x7fffffffffffLL; int bi = 0;
      for (int c = 0; c < 64; ++c) {
        if (used & (1ull << c)) continue;
        long long flat = (long long)ck[c] * NV + cs[c];
        if (cv[c] > best || (cv[c] == best && flat < bflat)) { best = cv[c]; bflat = flat; bi = c; }
      }
      used |= 1ull << bi;
      selv[r] = cv[bi]; selp[r] = ck[bi]; sels[r] = cs[bi];
    }
  }
  __syncthreads();
  if (tid < 8) {
    out[O_SCORES  + b * 8 + tid] = selv[tid];
    out[O_PARENTS + b * 8 + tid] = (float)selp[tid];
    ws_par[b * 8 + tid] = selp[tid];
  }
  for (int r = 0; r < 8; ++r) {
    const int* src = cand_seqs + ((size_t)b * 8 + selp[r]) * LSEQ;
    int dst = O_NEWSEQS + (b * 8 + r) * (LSEQ + 1);
    for (int t = tid; t <= LSEQ; t += 128) out[dst + t] = (float)(t < LSEQ ? src[t] : sels[r]);
  }
}

// ---------------------------------------------------------------------------
// Kernel 4: gather decoder state by parent (pure b128 streaming copy).
// One block per (b,k) destination beam.
// ---------------------------------------------------------------------------
__global__ __launch_bounds__(256) void k_gather(
    const float* __restrict__ ctx, const float* __restrict__ r1, const float* __restrict__ r2,
    const int* __restrict__ ws_par, float* __restrict__ out) {
  const int blk = blockIdx.x, tid = threadIdx.x;
  const int src = (blk >> 3) * 8 + ws_par[blk];
  {
    const float4* sp = (const float4*)(ctx + (size_t)src * ADIM);
    float4*       dp = (float4*)(out + O_CTX + (size_t)blk * ADIM);
    for (int i = tid; i < ADIM / 4; i += 256) dp[i] = sp[i];
  }
  {
    const float4* sp = (const float4*)(r1 + (size_t)src * DDIM);
    float4*       dp = (float4*)(out + O_R1 + (size_t)blk * DDIM);
    for (int i = tid; i < DDIM / 4; i += 256) dp[i] = sp[i];
  }
  {
    const float4* sp = (const float4*)(r2 + (size_t)src * DDIM);
    float4*       dp = (float4*)(out + O_R2 + (size_t)blk * DDIM);
    for (int i = tid; i < DDIM / 4; i += 256) dp[i] = sp[i];
  }
}

extern "C" void kernel_launch(void* const* d_in, const int* in_sizes, int n_in,
                              void* d_out, int out_size, void* d_ws, size_t ws_size,
                              hipStream_t stream) {
  (void)in_sizes; (void)n_in; (void)out_size; (void)ws_size;
  const float* logits           = (const float*)d_in[0];
  const float* cand_scores      = (const float*)d_in[1];
  const int*   cand_seqs        = (const int*)d_in[2];
  const float* completed_scores = (const float*)d_in[3];
  const int*   completed_seqs   = (const int*)d_in[4];
  const int*   completed_length = (const int*)d_in[5];
  const float* dctx             = (const float*)d_in[6];
  const float* dr1              = (const float*)d_in[7];
  const float* dr2              = (const float*)d_in[8];
  float* out = (float*)d_out;

  char* wsb   = (char*)d_ws;
  float* ws_eos = (float*)(wsb);                       // B*K floats
  float* ws_tv  = (float*)(wsb + 4096);                // B*K*8 floats
  int*   ws_ti  = (int*)  (wsb + 4096 + 32768);        // B*K*8 ints
  int*   ws_par = (int*)  (wsb + 4096 + 32768 + 32768);// B*K ints

  k_row_fused<<<NB * KB, 256, 0, stream>>>(logits, cand_scores, ws_eos, ws_tv, ws_ti);
  k_completed<<<NB, 128, 0, stream>>>(completed_scores, completed_seqs, completed_length,
                                      cand_seqs, ws_eos, out);
  k_newtop<<<NB, 128, 0, stream>>>(ws_tv, ws_ti, cand_seqs, ws_par, out);
  k_gather<<<NB * KB, 256, 0, stream>>>(dctx, dr1, dr2, ws_par, out);
}